// ramsey_MPNN_73598559584529
// MI455X (gfx1250) — compile-verified
//
#include <hip/hip_runtime.h>
#include <hip/hip_bf16.h>

#define NN 2048
#define EE 2096128.0f   // N*(N-1)/2 edges
#define BNEPS 1e-5f
#define SLOPE 0.01f

typedef __attribute__((ext_vector_type(2))) float v2f;
typedef __attribute__((ext_vector_type(8))) float v8f;

// ---------------------------------------------------------------------------
// GEMM via V_WMMA_F32_16X16X4_F32 (fp32 matrix pipe, full precision).
// All shapes/epilogue flags are compile-time -> fully unrolled WMMA chain.
// C[2048 x HC] = act( ((1+eps)*A0 + agg) @ W[K x HC] + bias ) + add
//   ACT: 0 = none, 1 = relu, 2 = leaky(0.01)
// One wave per 16x16 output tile; 8 waves (256 thr) per block.
// ---------------------------------------------------------------------------
template <int K, int HC, int ACT, bool HAS_AGG, bool HAS_EPS, bool HAS_BIAS, bool HAS_ADD>
__global__ void gin_gemm_wmma(const float* __restrict__ A0,
                              const float* __restrict__ A1,
                              const float* __restrict__ epsPtr,
                              const float* __restrict__ W,
                              const float* __restrict__ bias,
                              const float* __restrict__ addPtr,
                              float* __restrict__ Cout)
{
  constexpr int colTiles = HC / 16;
  constexpr int rowTilesPerBlock = 8 / colTiles;
  const int lane = threadIdx.x & 31;
  const int wave = threadIdx.x >> 5;
  const int colTile = wave % colTiles;
  const int rowTile = wave / colTiles;
  const int r0 = blockIdx.x * (rowTilesPerBlock * 16) + rowTile * 16;
  const int c0 = colTile * 16;
  const int m  = lane & 15;   // A row within tile / B col within tile
  const int kh = lane >> 4;   // K-half selector

  float ascale = 1.0f;
  if (HAS_EPS) ascale += *epsPtr;

  const float* arow = A0 + (size_t)(r0 + m) * K + 2 * kh;
  const float* grow = HAS_AGG ? (A1 + (size_t)(r0 + m) * K + 2 * kh) : nullptr;
  const float* wcol = W + c0 + m;

  v8f acc = {};
#pragma unroll
  for (int kb = 0; kb < K; kb += 4) {
    const int k0 = kb + 2 * kh;             // K index for a.x/b.x ; .y is k0+1
    v2f a;
    a.x = arow[kb] * ascale;
    a.y = arow[kb + 1] * ascale;
    if (HAS_AGG) {                          // fused "+ agg" (GIN aggregation)
      a.x += grow[kb];
      a.y += grow[kb + 1];
    }
    v2f b;
    b.x = wcol[(size_t)k0 * HC];
    b.y = wcol[(size_t)(k0 + 1) * HC];
    acc = __builtin_amdgcn_wmma_f32_16x16x4_f32(false, a, false, b,
                                                (short)0, acc, false, false);
  }

  const int col   = c0 + m;
  const int rbase = r0 + 8 * kh;
  const float bv = HAS_BIAS ? bias[col] : 0.0f;
#pragma unroll
  for (int g = 0; g < 8; ++g) {
    float v = acc[g] + bv;
    if (ACT == 1)      v = fmaxf(v, 0.0f);
    else if (ACT == 2) v = (v > 0.0f) ? v : SLOPE * v;
    const int row = rbase + g;
    if (HAS_ADD) v += addPtr[(size_t)row * HC + col];  // residual (+xinit)
    Cout[(size_t)row * HC + col] = v;
  }
}

// ---------------------------------------------------------------------------
// Exclusive prefix sum over the node dimension per feature column.
// agg[n][c] = sum_{s<n} h[s][c].  One block per column, 256 threads x 8 rows.
// ---------------------------------------------------------------------------
__global__ void prefix_scan_nodes(const float* __restrict__ h,
                                  float* __restrict__ agg, int W)
{
  const int c = blockIdx.x;
  const int t = threadIdx.x;
  float vals[8];
  float s = 0.0f;
  const int base = t * 8;
  for (int i = 0; i < 8; ++i) {
    vals[i] = h[(size_t)(base + i) * W + c];
    s += vals[i];
  }
  __shared__ float sh[256];
  sh[t] = s;
  __syncthreads();
  for (int off = 1; off < 256; off <<= 1) {   // Hillis-Steele inclusive scan
    float v = (t >= off) ? sh[t - off] : 0.0f;
    __syncthreads();
    sh[t] += v;
    __syncthreads();
  }
  float run = sh[t] - s;                      // exclusive chunk offset
  for (int i = 0; i < 8; ++i) {
    agg[(size_t)(base + i) * W + c] = run;
    run += vals[i];
  }
}

// ---------------------------------------------------------------------------
// BatchNorm1d (training mode, biased stats) over the 2048 nodes, in place.
// One block per channel.
// ---------------------------------------------------------------------------
__global__ void bn_nodes(float* __restrict__ h,
                         const float* __restrict__ g,
                         const float* __restrict__ b, int W)
{
  const int c = blockIdx.x;
  const int t = threadIdx.x;
  float vals[8];
  float s = 0.0f, q = 0.0f;
  for (int i = 0; i < 8; ++i) {
    float v = h[(size_t)(t + i * 256) * W + c];
    vals[i] = v; s += v; q += v * v;
  }
  __shared__ float ss[256], sq[256];
  ss[t] = s; sq[t] = q;
  __syncthreads();
  for (int off = 128; off > 0; off >>= 1) {
    if (t < off) { ss[t] += ss[t + off]; sq[t] += sq[t + off]; }
    __syncthreads();
  }
  __shared__ float smean, srs;
  if (t == 0) {
    float mean = ss[0] * (1.0f / NN);
    float var  = sq[0] * (1.0f / NN) - mean * mean;
    smean = mean;
    srs   = rsqrtf(var + BNEPS);
  }
  __syncthreads();
  const float gamma = g[c], beta = b[c], mean = smean, rs = srs;
  for (int i = 0; i < 8; ++i)
    h[(size_t)(t + i * 256) * W + c] = (vals[i] - mean) * rs * gamma + beta;
}

// ---------------------------------------------------------------------------
// Edge phase pass 1: per-channel sum & sumsq of ep = leaky(A[s]+B[d]) over all
// edges (s<d), for the edge BatchNorm.  Block = one destination d.
// ---------------------------------------------------------------------------
__global__ void edge_stats(const float* __restrict__ A,
                           const float* __restrict__ B,
                           float* __restrict__ esum,
                           float* __restrict__ esq)
{
  const int d = blockIdx.x;
  const int t = threadIdx.x;
  const int c = t & 63;
  const int j = t >> 6;
  const float Bd = B[(size_t)d * 64 + c];
  float s = 0.0f, q = 0.0f;
  for (int si = j; si < d; si += 4) {
    float v = A[(size_t)si * 64 + c] + Bd;
    v = (v > 0.0f) ? v : SLOPE * v;
    s += v; q += v * v;
  }
  __shared__ float sh[256], shq[256];
  sh[t] = s; shq[t] = q;
  __syncthreads();
  if (j == 0) {
    s = sh[c] + sh[c + 64] + sh[c + 128] + sh[c + 192];
    q = shq[c] + shq[c + 64] + shq[c + 128] + shq[c + 192];
    atomicAdd(&esum[c], s);
    atomicAdd(&esq[c], q);
  }
}

__global__ void zero_stats(float* esum, float* esq)
{
  const int t = threadIdx.x;
  if (t < 64) { esum[t] = 0.0f; esq[t] = 0.0f; }
}

// ---------------------------------------------------------------------------
// Fold edge-BN (scale/shift) into lin6:  W6f[c][k] = scale[c]*W6[c][k],
// b6f[k] = b6[k] + sum_c shift[c]*W6[c][k].
// ---------------------------------------------------------------------------
__global__ void fold_bn_lin6(const float* __restrict__ esum,
                             const float* __restrict__ esq,
                             const float* __restrict__ g5,
                             const float* __restrict__ b5,
                             const float* __restrict__ W6,
                             const float* __restrict__ b6,
                             float* __restrict__ W6f,
                             float* __restrict__ b6f)
{
  const int t = threadIdx.x;  // 128 threads
  __shared__ float scale[64], shift[64];
  if (t < 64) {
    const float invE = 1.0f / EE;
    float mean = esum[t] * invE;
    float var  = esq[t] * invE - mean * mean;
    float rs   = rsqrtf(var + BNEPS);
    float sc   = g5[t] * rs;
    scale[t] = sc;
    shift[t] = b5[t] - mean * sc;
  }
  __syncthreads();
  W6f[t] = scale[t >> 1] * W6[t];           // t = c*2 + k
  if (t < 2) {
    float acc = b6[t];
    for (int c = 0; c < 64; ++c) acc += shift[c] * W6[c * 2 + t];
    b6f[t] = acc;
  }
}

// ---------------------------------------------------------------------------
// Edge phase pass 2: recompute ep, fused BN+lin6, softmax(2), symmetric
// scatter into probs[N][N][2].  Block = one destination d; thread = source s.
// ---------------------------------------------------------------------------
__global__ void edge_emit(const float* __restrict__ A,
                          const float* __restrict__ B,
                          const float* __restrict__ W6f,
                          const float* __restrict__ b6f,
                          float* __restrict__ out)
{
  const int d = blockIdx.x;
  const int t = threadIdx.x;
  __shared__ float Bd[64];
  __shared__ float sW[128];
  __shared__ float sb[2];
  if (t < 64)  Bd[t] = B[(size_t)d * 64 + t];
  if (t < 128) sW[t] = W6f[t];
  if (t < 2)   sb[t] = b6f[t];
  __syncthreads();

  float2* out2 = (float2*)out;              // C=2 packs into float2
  for (int s = t; s < d; s += 256) {
    const float4* a4 = (const float4*)(A + (size_t)s * 64);
    // prefetch next chunk's A row (global_prefetch_b8; speculative, safe)
    __builtin_prefetch(A + (size_t)(s + 256) * 64, 0, 0);
    float acc0 = sb[0], acc1 = sb[1];
#pragma unroll
    for (int c4 = 0; c4 < 16; ++c4) {
      float4 av = a4[c4];
      const int c = c4 * 4;
      float v;
      v = av.x + Bd[c + 0]; v = (v > 0.0f) ? v : SLOPE * v;
      acc0 += v * sW[(c + 0) * 2]; acc1 += v * sW[(c + 0) * 2 + 1];
      v = av.y + Bd[c + 1]; v = (v > 0.0f) ? v : SLOPE * v;
      acc0 += v * sW[(c + 1) * 2]; acc1 += v * sW[(c + 1) * 2 + 1];
      v = av.z + Bd[c + 2]; v = (v > 0.0f) ? v : SLOPE * v;
      acc0 += v * sW[(c + 2) * 2]; acc1 += v * sW[(c + 2) * 2 + 1];
      v = av.w + Bd[c + 3]; v = (v > 0.0f) ? v : SLOPE * v;
      acc0 += v * sW[(c + 3) * 2]; acc1 += v * sW[(c + 3) * 2 + 1];
    }
    const float mx = fmaxf(acc0, acc1);
    const float e0 = __expf(acc0 - mx);
    const float e1 = __expf(acc1 - mx);
    const float inv = 1.0f / (e0 + e1);
    float2 p; p.x = e0 * inv; p.y = e1 * inv;
    out2[(size_t)s * NN + d] = p;           // probs[s,d]
    out2[(size_t)d * NN + s] = p;           // probs[d,s]
  }
}

__global__ void zero_diag(float* __restrict__ out)
{
  const int i = blockIdx.x * 256 + threadIdx.x;
  if (i < NN) {
    float2 z; z.x = 0.0f; z.y = 0.0f;
    ((float2*)out)[(size_t)i * NN + i] = z;
  }
}

// ---------------------------------------------------------------------------
template <int K, int HC, int ACT, bool HAS_AGG, bool HAS_EPS, bool HAS_BIAS, bool HAS_ADD>
static void launch_gemm(const float* A0, const float* A1, const float* eps,
                        const float* W, const float* bias, const float* add,
                        float* C, hipStream_t stream)
{
  constexpr int colTiles = HC / 16;
  constexpr int rowTilesPerBlock = 8 / colTiles;
  constexpr int blocks = NN / (16 * rowTilesPerBlock);
  gin_gemm_wmma<K, HC, ACT, HAS_AGG, HAS_EPS, HAS_BIAS, HAS_ADD>
      <<<blocks, 256, 0, stream>>>(A0, A1, eps, W, bias, add, C);
}

extern "C" void kernel_launch(void* const* d_in, const int* in_sizes, int n_in,
                              void* d_out, int out_size, void* d_ws, size_t ws_size,
                              hipStream_t stream)
{
  (void)in_sizes; (void)n_in; (void)out_size; (void)ws_size;
  const float* node_features = (const float*)d_in[1];      // xinit and h0
  const float* conv1_W1 = (const float*)d_in[2];
  const float* conv1_b1 = (const float*)d_in[3];
  const float* conv1_W2 = (const float*)d_in[4];
  const float* conv1_b2 = (const float*)d_in[5];
  const float* conv1_eps = (const float*)d_in[6];
  const float* convs_W1 = (const float*)d_in[7];
  const float* convs_b1 = (const float*)d_in[8];
  const float* convs_W2 = (const float*)d_in[9];
  const float* convs_b2 = (const float*)d_in[10];
  const float* convs_eps = (const float*)d_in[11];
  const float* bns_gamma = (const float*)d_in[12];
  const float* bns_beta = (const float*)d_in[13];
  const float* lin1_W = (const float*)d_in[14];
  const float* lin1_b = (const float*)d_in[15];
  const float* bn1_g = (const float*)d_in[16];
  const float* bn1_b = (const float*)d_in[17];
  const float* lin2_W = (const float*)d_in[18];
  const float* lin2_b = (const float*)d_in[19];
  const float* bn2_g = (const float*)d_in[20];
  const float* bn2_b = (const float*)d_in[21];
  const float* lin3_W = (const float*)d_in[22];
  const float* lin3_b = (const float*)d_in[23];
  const float* lin5_W = (const float*)d_in[24];  // [32,64]
  const float* lin5_b = (const float*)d_in[25];
  const float* bn5_g = (const float*)d_in[26];
  const float* bn5_b = (const float*)d_in[27];
  const float* lin6_W = (const float*)d_in[28];  // [64,2]
  const float* lin6_b = (const float*)d_in[29];

  float* ws = (float*)d_ws;                 // ~2.8 MB total
  float* hA    = ws;             ws += NN * 64;
  float* hB    = ws;             ws += NN * 64;
  float* agg   = ws;             ws += NN * 64;
  float* Anode = ws;             ws += NN * 64;
  float* Bnode = ws;             ws += NN * 64;
  float* hout  = ws;             ws += NN * 16;
  float* esum  = ws;             ws += 64;
  float* esq   = ws;             ws += 64;
  float* W6f   = ws;             ws += 128;
  float* b6f   = ws;             ws += 2;

  // --- conv1 (GIN F->H->H), leaky(relu(x)) == relu(x) ---
  prefix_scan_nodes<<<16, 256, 0, stream>>>(node_features, agg, 16);
  launch_gemm<16, 64, 1, true, true, true, false>(
      node_features, agg, conv1_eps, conv1_W1, conv1_b1, nullptr, hB, stream);
  launch_gemm<64, 64, 1, false, false, true, false>(
      hB, nullptr, nullptr, conv1_W2, conv1_b2, nullptr, hA, stream);

  // --- extra GIN layers + BN ---
  for (int l = 0; l < 2; ++l) {
    prefix_scan_nodes<<<64, 256, 0, stream>>>(hA, agg, 64);
    launch_gemm<64, 64, 1, true, true, true, false>(
        hA, agg, convs_eps + l, convs_W1 + l * 64 * 64, convs_b1 + l * 64,
        nullptr, hB, stream);
    launch_gemm<64, 64, 1, false, false, true, false>(
        hB, nullptr, nullptr, convs_W2 + l * 64 * 64, convs_b2 + l * 64,
        nullptr, hA, stream);
    bn_nodes<<<64, 256, 0, stream>>>(hA, bns_gamma + l * 64, bns_beta + l * 64, 64);
  }

  // --- head MLP: BN(leaky(lin1)), BN(leaky(lin2)), lin3 + xinit ---
  launch_gemm<64, 64, 2, false, false, true, false>(
      hA, nullptr, nullptr, lin1_W, lin1_b, nullptr, hB, stream);
  bn_nodes<<<64, 256, 0, stream>>>(hB, bn1_g, bn1_b, 64);
  launch_gemm<64, 64, 2, false, false, true, false>(
      hB, nullptr, nullptr, lin2_W, lin2_b, nullptr, hA, stream);
  bn_nodes<<<64, 256, 0, stream>>>(hA, bn2_g, bn2_b, 64);
  launch_gemm<64, 16, 0, false, false, true, true>(
      hA, nullptr, nullptr, lin3_W, lin3_b, node_features, hout, stream);

  // --- edge factorization: A = hout@W5_top, B = hout@W5_bot + b5 ---
  launch_gemm<16, 64, 0, false, false, false, false>(
      hout, nullptr, nullptr, lin5_W, nullptr, nullptr, Anode, stream);
  launch_gemm<16, 64, 0, false, false, true, false>(
      hout, nullptr, nullptr, lin5_W + 16 * 64, lin5_b, nullptr, Bnode, stream);

  // --- edge BN stats -> fold into lin6 -> emit probs ---
  zero_stats<<<1, 64, 0, stream>>>(esum, esq);
  edge_stats<<<NN, 256, 0, stream>>>(Anode, Bnode, esum, esq);
  fold_bn_lin6<<<1, 128, 0, stream>>>(esum, esq, bn5_g, bn5_b, lin6_W, lin6_b, W6f, b6f);
  edge_emit<<<NN, 256, 0, stream>>>(Anode, Bnode, W6f, b6f, (float*)d_out);
  zero_diag<<<(NN + 255) / 256, 256, 0, stream>>>((float*)d_out);
}